// DecodePredictions_4672924418594
// MI455X (gfx1250) — compile-verified
//
#include <hip/hip_runtime.h>
#include <hip/hip_bf16.h>
#include <math.h>

// ---------------- problem constants (match reference) ----------------
#define BATCH   8
#define NANCH   49104      // total anchors for 512x512
#define NCLS    80
#define KPC     100        // K_PER_CLASS
#define MAXDET  100
#define CONF_LOGIT  (-2.9444389792f)   // logit(0.05)
#define IOU_T   0.5f
#define NEG_SENT (-1.0e30f)            // sentinel for suppressed (logit domain)

typedef __attribute__((ext_vector_type(2))) float    v2f;
typedef __attribute__((ext_vector_type(8))) float    v8f;
typedef __attribute__((ext_vector_type(4))) unsigned v4u;
typedef __attribute__((ext_vector_type(8))) unsigned v8u;

#define USE_ASYNC 1
#define USE_TDM   1

// gfx1250 async global->LDS copy (ASYNCcnt-tracked). LDS address is the raw
// 32-bit LDS byte offset (low 32 bits of a generic pointer to __shared__).
__device__ __forceinline__ void async_ld_b128(unsigned lds_off, unsigned long long gaddr) {
#if USE_ASYNC
  asm volatile("global_load_async_to_lds_b128 %0, %1, off"
               :: "v"(lds_off), "v"(gaddr) : "memory");
#endif
}
__device__ __forceinline__ void wait_async0() {
#if USE_ASYNC
  asm volatile("s_wait_asynccnt 0" ::: "memory");
#endif
}

__device__ __forceinline__ unsigned f2key(float f) {   // order-preserving f32->u32
  unsigned b = __float_as_uint(f);
  return (b & 0x80000000u) ? ~b : (b | 0x80000000u);
}
__device__ __forceinline__ float key2f(unsigned k) {
  unsigned b = (k & 0x80000000u) ? (k ^ 0x80000000u) : ~k;
  return __uint_as_float(b);
}

// ============================================================================
// Kernel 1: decode boxes + transpose class logits (one streaming pass).
// Block = 64 anchors of one image; 64*84 floats staged in LDS via async copy.
// ============================================================================
__global__ __launch_bounds__(256) void k_decode_transpose(
    const float* __restrict__ pred, float* __restrict__ boxesWs,
    float* __restrict__ scoresT)
{
  __shared__ float tile[64 * 84];               // 21.5 KB
  const int tid = threadIdx.x;
  const int b  = blockIdx.y;
  const int n0 = blockIdx.x * 64;
  const int nvalid = min(64, NANCH - n0);
  const float* src = pred + ((size_t)b * NANCH + n0) * 84;
  const int nf4 = 21 * nvalid;                  // 84*nvalid floats / 4, exact

#if USE_ASYNC
  {
    unsigned ldsBase = (unsigned)(size_t)&tile[0];
    unsigned long long g = (unsigned long long)(size_t)src;
    for (int e = tid; e < nf4; e += 256)
      async_ld_b128(ldsBase + (unsigned)e * 16u, g + (unsigned long long)e * 16u);
  }
  if (blockIdx.x + 2 < gridDim.x) {             // prefetch next chunk into L2
    const float* nxt = src + 64 * 84;
    if (tid < 168) __builtin_prefetch(nxt + tid * 32, 0, 1);
  }
  wait_async0();
#else
  for (int e = tid; e < nf4; e += 256)
    ((float4*)tile)[e] = ((const float4*)src)[e];
#endif
  __syncthreads();

  // ---- decode 64 boxes (anchor computed on the fly) ----
  if (tid < 64 && (n0 + tid) < NANCH) {
    int n = n0 + tid;
    int lvl, base;
    if      (n < 36864) { lvl = 0; base = 0; }
    else if (n < 46080) { lvl = 1; base = 36864; }
    else if (n < 48384) { lvl = 2; base = 46080; }
    else if (n < 48960) { lvl = 3; base = 48384; }
    else                { lvl = 4; base = 48960; }
    int idx  = n - base;
    int cell = idx / 9;
    int a    = idx - cell * 9;
    int fw   = 64 >> lvl;
    int gx   = cell & (fw - 1);
    int gy   = cell >> (6 - lvl);
    float stride = (float)(8 << lvl);
    float side   = (float)(32 << lvl);
    const float SQ[3] = {0.70710678118654752f, 1.0f, 1.41421356237309505f}; // sqrt(r)
    const float SC[3] = {1.0f, 1.2599210498948732f, 1.5874010519681994f};   // 2^(s/3)
    int ri = a / 3, si = a - ri * 3;
    float aw = side * SQ[ri] * SC[si];
    float ah = side / SQ[ri] * SC[si];
    float cx = ((float)gx + 0.5f) * stride;
    float cy = ((float)gy + 0.5f) * stride;
    const float* p = &tile[tid * 84];
    float bxc = p[0] * 0.1f * aw + cx;
    float byc = p[1] * 0.1f * ah + cy;
    float bw  = expf(p[2] * 0.2f) * aw;
    float bh  = expf(p[3] * 0.2f) * ah;
    float4 box = make_float4(bxc - 0.5f * bw, byc - 0.5f * bh,
                             bxc + 0.5f * bw, byc + 0.5f * bh);
    ((float4*)boxesWs)[(size_t)b * NANCH + n] = box;
  }

  // ---- transposed class-logit writes (coalesced 64-float runs per class) ----
  for (int e = tid; e < 64 * NCLS; e += 256) {
    int c = e >> 6, j = e & 63;
    int n = n0 + j;
    if (n < NANCH)
      scoresT[((size_t)(b * NCLS + c)) * NANCH + n] = tile[j * 84 + 4 + c];
  }
}

// ============================================================================
// Kernel 2: exact top-100 per (image,class). Whole 49104-float column staged
// in LDS (196 KB of the WGP's 320 KB) by ONE Tensor-Data-Mover descriptor,
// then a 32-step binary search on u32 keys; candidates bitonic-sorted.
// ============================================================================
__global__ __launch_bounds__(256) void k_topk(
    const float* __restrict__ scoresT, int* __restrict__ topkIdx,
    float* __restrict__ clsLogit)
{
  extern __shared__ unsigned keys[];            // 49152 u32 = 196,608 B dynamic
  __shared__ int scnt;
  __shared__ int candn;
  __shared__ unsigned long long cand[128];
  const int tid = threadIdx.x;
  const int b = blockIdx.y, c = blockIdx.x;
  const float* col = scoresT + (size_t)(b * NCLS + c) * NANCH;

#if USE_TDM
  // One TDM descriptor: 1-row tile of 49104 dwords, contiguous -> LDS.
  if (tid < 32) {                               // wave 0 only (TENSORcnt per-wave)
    unsigned long long g = (unsigned long long)(size_t)col;
    unsigned ldsAddr = (unsigned)(size_t)&keys[0];
    v4u g0;
    g0[0] = 1u;                                 // count=1 (valid user D#)
    g0[1] = ldsAddr;                            // lds_addr (bytes)
    g0[2] = (unsigned)(g & 0xFFFFFFFFu);        // global_addr[31:0]
    g0[3] = (unsigned)(g >> 32) | 0x80000000u;  // global_addr[56:32] | type=2
    v8u g1;
    g1[0] = 2u << 16;                           // data_size=4B; no multicast/pad
    g1[1] = ((unsigned)NANCH & 0xFFFFu) << 16;  // tensor_dim0[15:0] @bits63:48
    g1[2] = (((unsigned)NANCH >> 16) & 0xFFFFu) // tensor_dim0[31:16]
          | (1u << 16);                         // tensor_dim1[15:0] = 1
    g1[3] = ((unsigned)NANCH & 0xFFFFu) << 16;  // tensor_dim1 hi=0 | tile_dim0
    g1[4] = 1u;                                 // tile_dim1=1, tile_dim2=0
    g1[5] = (unsigned)NANCH;                    // tensor_dim0_stride[31:0]
    g1[6] = 0u;                                 // stride0 hi | stride1 lo
    g1[7] = 0u;                                 // stride1 hi
    asm volatile("tensor_load_to_lds %0, %1" :: "s"(g0), "s"(g1) : "memory");
    __builtin_amdgcn_s_wait_tensorcnt(0);
  }
#elif USE_ASYNC
  {
    unsigned ldsBase = (unsigned)(size_t)&keys[0];
    unsigned long long g = (unsigned long long)(size_t)col;
    for (int e = tid; e < NANCH / 4; e += 256)   // 12276 b128 transfers, aligned
      async_ld_b128(ldsBase + (unsigned)e * 16u, g + (unsigned long long)e * 16u);
  }
  wait_async0();
#else
  for (int e = tid; e < NANCH / 4; e += 256)
    ((float4*)keys)[e] = ((const float4*)col)[e];
#endif
  __syncthreads();

  // in-place transform to order-preserving u32; pad tail with 0 (= min key)
  for (int e = tid; e < 49152; e += 256) {
    unsigned k = 0u;
    if (e < NANCH) {
      unsigned bits = keys[e];
      k = (bits & 0x80000000u) ? ~bits : (bits | 0x80000000u);
    }
    keys[e] = k;
  }
  if (tid == 0) candn = 0;
  __syncthreads();

  // find max threshold t with count(key >= t) >= KPC
  unsigned lo = 0u, hi = 0xFFFFFFFFu;
  while (lo < hi) {
    unsigned mid = (unsigned)(((unsigned long long)lo + hi + 1ull) >> 1);
    if (tid == 0) scnt = 0;
    __syncthreads();
    int local = 0;
    for (int e = tid; e < 49152; e += 256) local += (keys[e] >= mid) ? 1 : 0;
    atomicAdd(&scnt, local);
    __syncthreads();
    int cnt = scnt;
    __syncthreads();
    if (cnt >= KPC) lo = mid; else hi = mid - 1;
  }
  const unsigned thr = lo;

  // collect candidates (>= thr; always >= 100 of them, capped at 128)
  for (int e = tid; e < NANCH; e += 256) {
    unsigned k = keys[e];
    if (k >= thr) {
      int pos = atomicAdd(&candn, 1);
      if (pos < 128)
        cand[pos] = ~((((unsigned long long)k) << 32) | (unsigned)(~e));
    }
  }
  __syncthreads();
  int nc = min(candn, 128);
  if (tid < 128 && tid >= nc) cand[tid] = ~0ull;  // pads sort last
  __syncthreads();

  // ascending bitonic sort of 128 keys (=> best candidate first)
  for (int k = 2; k <= 128; k <<= 1)
    for (int j = k >> 1; j > 0; j >>= 1) {
      if (tid < 128) {
        int i = tid, l = i ^ j;
        if (l > i) {
          bool asc = ((i & k) == 0);
          unsigned long long A = cand[i], B2 = cand[l];
          if ((A > B2) == asc) { cand[i] = B2; cand[l] = A; }
        }
      }
      __syncthreads();
    }

  if (tid < KPC) {
    unsigned long long comp = ~cand[tid];
    unsigned tk = (unsigned)(comp >> 32);
    int idx = (int)(~(unsigned)comp);
    idx = min(max(idx, 0), NANCH - 1);
    size_t o = (size_t)(b * NCLS + c) * KPC + tid;
    topkIdx[o] = idx;
    clsLogit[o] = key2f(tk);
  }
}

// ============================================================================
// Kernel 3: per-(image,class) NMS. One wave32 (EXEC all-ones for WMMA).
// Union base (area_i + area_j) for each 16x16 IoU tile is computed with
// v_wmma_f32_16x16x4_f32 as a rank-2 outer product; intersection on VALU.
// Division-free threshold: inter/max(u,eps) > T  <=>  inter > T*max(u,eps).
// ============================================================================
__global__ __launch_bounds__(32) void k_nms(
    const float* __restrict__ boxesWs, const int* __restrict__ topkIdx,
    float* __restrict__ clsLogit, float* __restrict__ candBoxes)
{
  __shared__ float bxs[112][4];
  __shared__ float area[112];
  __shared__ float lgs[112];
  __shared__ unsigned msk[112][4];   // suppression bitmask rows
  __shared__ unsigned kp[4];         // keep bits
  const int lane = threadIdx.x;
  const int b = blockIdx.y, c = blockIdx.x;
  const size_t base = (size_t)(b * NCLS + c) * KPC;

  for (int t = lane; t < 112; t += 32) {
    float4 box = make_float4(-1e30f, -1e30f, -1e30f, -1e30f);
    float l = NEG_SENT;
    if (t < KPC) {
      int idx = topkIdx[base + t];
      l = clsLogit[base + t];
      box = ((const float4*)boxesWs)[(size_t)b * NANCH + idx];
      ((float4*)candBoxes)[base + t] = box;   // reference keeps boxes for all K
    }
    bxs[t][0] = box.x; bxs[t][1] = box.y; bxs[t][2] = box.z; bxs[t][3] = box.w;
    area[t] = (box.z - box.x) * (box.w - box.y);
    lgs[t] = l;
  }
  for (int e = lane; e < 112 * 4; e += 32) ((unsigned*)msk)[e] = 0u;
  __syncthreads();
  if (lane < 4) {
    unsigned k0 = 0u;
    for (int i2 = 0; i2 < 32; i2++) {
      int t = (lane << 5) + i2;
      if (t < KPC && lgs[t] > CONF_LOGIT) k0 |= (1u << i2);
    }
    kp[lane] = k0;
  }
  __syncthreads();

  const int half = lane >> 4;        // wave32 operand layout halves
  const int ln   = lane & 15;
  for (int ti = 0; ti < 7; ti++) {
    for (int tj = 0; tj < 7; tj++) {
      // A(16x4): col0 = area_i, col1 = 1, cols2-3 = 0 (lanes 16-31 hold K=2,3)
      // B(4x16): row0 = 1, row1 = area_j, rows2-3 = 0 (lanes 16-31 hold K=2,3)
      v2f A, B;
      if (half == 0) {
        A.x = area[ti * 16 + ln]; A.y = 1.0f;
        B.x = 1.0f;               B.y = area[tj * 16 + ln];
      } else {
        A.x = 0.0f; A.y = 0.0f;   B.x = 0.0f; B.y = 0.0f;
      }
      v8f C = {0.f, 0.f, 0.f, 0.f, 0.f, 0.f, 0.f, 0.f};
      v8f D = __builtin_amdgcn_wmma_f32_16x16x4_f32(
          false, A, false, B, (short)0, C, false, false);
#pragma unroll
      for (int r = 0; r < 8; r++) {
        int M  = ti * 16 + r + (half << 3);
        int Nc = tj * 16 + ln;
        float ltx = fmaxf(bxs[M][0], bxs[Nc][0]);
        float lty = fmaxf(bxs[M][1], bxs[Nc][1]);
        float rbx = fminf(bxs[M][2], bxs[Nc][2]);
        float rby = fminf(bxs[M][3], bxs[Nc][3]);
        float inter = fmaxf(rbx - ltx, 0.f) * fmaxf(rby - lty, 0.f);
        float uni = fmaxf(D[r] - inter, 1e-8f);
        // division-free:  inter/uni > IOU_T  <=>  inter > IOU_T*uni  (uni > 0)
        if (M < KPC && Nc < KPC && inter > IOU_T * uni)
          atomicOr(&msk[M][Nc >> 5], 1u << (Nc & 31));
      }
    }
  }
  __syncthreads();

  if (lane == 0) {                   // sequential suppression (matches fori_loop)
    unsigned kw[4] = {kp[0], kp[1], kp[2], kp[3]};
    for (int i = 0; i < KPC; i++) {
      if ((kw[i >> 5] >> (i & 31)) & 1u) {
        int wi = i >> 5, bi = i & 31;
        unsigned lowmask = (bi == 31) ? 0xFFFFFFFFu : ((2u << bi) - 1u);
        for (int w = 0; w < 4; w++) {
          unsigned m = msk[i][w];
          if (w < wi) m = 0u;
          else if (w == wi) m &= ~lowmask;   // only suppress j > i
          kw[w] &= ~m;
        }
      }
    }
    kp[0] = kw[0]; kp[1] = kw[1]; kp[2] = kw[2]; kp[3] = kw[3];
  }
  __syncthreads();
  for (int t = lane; t < KPC; t += 32) {
    bool kept = (kp[t >> 5] >> (t & 31)) & 1u;
    clsLogit[base + t] = kept ? lgs[t] : NEG_SENT;
  }
}

// ============================================================================
// Kernel 4: per-image global top-100 over 80*100 entries via 8192-key bitonic
// sort in 64 KB LDS; emit boxes/scores/classes/valid.
// ============================================================================
__global__ __launch_bounds__(256) void k_final(
    const float* __restrict__ clsLogit, const float* __restrict__ candBoxes,
    float* __restrict__ outBoxes, float* __restrict__ outScores,
    float* __restrict__ outClasses, float* __restrict__ outValid)
{
  extern __shared__ unsigned long long srt[];   // 8192 * 8B dynamic
  __shared__ int vcnt;
  const int tid = threadIdx.x;
  const int b = blockIdx.x;
  const float* lgB = clsLogit + (size_t)b * NCLS * KPC;

  for (int e = tid; e < 8192; e += 256) {
    unsigned long long sk = ~0ull;
    if (e < NCLS * KPC) {
      unsigned tk = f2key(lgB[e]);
      sk = ~((((unsigned long long)tk) << 32) | (unsigned)(~e));
    }
    srt[e] = sk;
  }
  if (tid == 0) vcnt = 0;
  __syncthreads();

  for (int k = 2; k <= 8192; k <<= 1)
    for (int j = k >> 1; j > 0; j >>= 1) {
      for (int i = tid; i < 8192; i += 256) {
        int l = i ^ j;
        if (l > i) {
          bool asc = ((i & k) == 0);
          unsigned long long A = srt[i], B2 = srt[l];
          if ((A > B2) == asc) { srt[i] = B2; srt[l] = A; }
        }
      }
      __syncthreads();
    }

  if (tid < MAXDET) {
    unsigned long long comp = ~srt[tid];
    int flat = (int)(~(unsigned)comp);
    flat = min(max(flat, 0), NCLS * KPC - 1);
    float lgv = lgB[flat];
    bool mask = lgv > -1e29f;                       // kept (reference: fs > 0)
    float sc = mask ? (1.0f / (1.0f + expf(-lgv))) : 0.0f;
    float4 box = make_float4(0.f, 0.f, 0.f, 0.f);
    if (mask) box = ((const float4*)candBoxes)[(size_t)b * NCLS * KPC + flat];
    ((float4*)outBoxes)[b * MAXDET + tid] = box;
    outScores[b * MAXDET + tid] = sc;
    outClasses[b * MAXDET + tid] = mask ? (float)(flat / KPC) : 0.0f;
    if (mask) atomicAdd(&vcnt, 1);
  }
  __syncthreads();
  if (tid == 0) outValid[b] = (float)vcnt;
}

// ============================================================================
extern "C" void kernel_launch(void* const* d_in, const int* in_sizes, int n_in,
                              void* d_out, int out_size, void* d_ws, size_t ws_size,
                              hipStream_t stream) {
  (void)in_sizes; (void)n_in; (void)out_size; (void)ws_size;
  const float* pred = (const float*)d_in[1];          // predictions [8,49104,84]

  // workspace layout (all 16B aligned): ~133.5 MB total
  float* boxesWs   = (float*)d_ws;                                  // B*N*4
  float* scoresT   = boxesWs + (size_t)BATCH * NANCH * 4;           // B*C*N
  int*   topkIdx   = (int*)(scoresT + (size_t)BATCH * NCLS * NANCH);// B*C*K
  float* clsLogit  = (float*)(topkIdx + BATCH * NCLS * KPC);        // B*C*K
  float* candBoxes = clsLogit + BATCH * NCLS * KPC;                 // B*C*K*4

  float* outBoxes   = (float*)d_out;                  // [8,100,4]
  float* outScores  = outBoxes  + BATCH * MAXDET * 4; // [8,100]
  float* outClasses = outScores + BATCH * MAXDET;     // [8,100]
  float* outValid   = outClasses + BATCH * MAXDET;    // [8]

  dim3 g1((NANCH + 63) / 64, BATCH);
  k_decode_transpose<<<g1, 256, 0, stream>>>(pred, boxesWs, scoresT);

  dim3 g2(NCLS, BATCH);
  k_topk<<<g2, 256, 49152 * sizeof(unsigned), stream>>>(scoresT, topkIdx, clsLogit);
  k_nms<<<g2, 32, 0, stream>>>(boxesWs, topkIdx, clsLogit, candBoxes);

  k_final<<<BATCH, 256, 8192 * sizeof(unsigned long long), stream>>>(
      clsLogit, candBoxes, outBoxes, outScores, outClasses, outValid);
}